// GNNModel_80487687127268
// MI455X (gfx1250) — compile-verified
//
#include <hip/hip_runtime.h>

// GCN (2-layer, PyG GCNConv semantics) for MI455X / gfx1250.
// fp32 throughout; GEMMs use V_WMMA_F32_16X16X4_F32 (wave32 matrix pipe),
// W staged to LDS via GLOBAL_LOAD_ASYNC_TO_LDS_B32 (ASYNCcnt-tracked).

typedef float v2f __attribute__((ext_vector_type(2)));
typedef float v8f __attribute__((ext_vector_type(8)));

#define TPB 256

__global__ void k_fill(float* __restrict__ p, float val, int n) {
    int i = blockIdx.x * blockDim.x + threadIdx.x;
    if (i < n) p[i] = val;
}

__global__ void k_deg_count(const int* __restrict__ dst, float* __restrict__ deg, int E) {
    int e = blockIdx.x * blockDim.x + threadIdx.x;
    if (e < E) {
        __hip_atomic_fetch_add(&deg[dst[e]], 1.0f,
                               __ATOMIC_RELAXED, __HIP_MEMORY_SCOPE_AGENT);
    }
}

__global__ void k_rsqrt(float* __restrict__ p, int n) {
    int i = blockIdx.x * blockDim.x + threadIdx.x;
    if (i < n) p[i] = rsqrtf(p[i]);
}

// H[N x 64] = op(A)[N x K] @ Wm[K x 64], op = identity or ReLU (fused on A loads).
// Also writes AGG = H * dinv[row]^2 + bias[col] (self-loop + bias, full overwrite).
// One wave owns a 16-row stripe across all 64 output columns (4 WMMA tiles),
// reusing the A fragment 4x per k-step.
// Fragment layouts per CDNA5 ISA (16x16x4 f32), with g = lane>=16, c = lane&15:
//   A (16x4):  vgpr v holds A[row=c, k + v + 2g]       -> contiguous b64 load
//   B (4x16):  vgpr v holds W[k + v + 2g, col=c]       -> one b64 LDS load from
//              pair-interleaved staging lwp[(kb/2)*64 + c] = {W[kb][c], W[kb+1][c]}
//   C (16x16): vgpr v holds D[row = v + 8g, col = c]
template <bool RELU, int K>
__global__ __launch_bounds__(TPB)
void k_gemm_wmma(const float* __restrict__ A, const float* __restrict__ Wm,
                 float* __restrict__ H, float* __restrict__ AGG,
                 const float* __restrict__ dinv, const float* __restrict__ bias,
                 int N)
{
    constexpr int F = 64;
    __shared__ v2f lwp[(K / 2) * F];   // K-pair interleaved copy of W

    const int tid = threadIdx.x;

    // Async global->LDS staging with on-the-fly pair interleave:
    // global element i = W[k][f] -> LDS float slot ((k/2)*64 + f)*2 + (k&1).
    {
        float* lds_f = (float*)lwp;
        for (int i = tid; i < K * F; i += TPB) {   // uniform trip count: EXEC full
            const int k = i >> 6;
            const int f = i & 63;
            const int s = ((((k >> 1) * F) + f) << 1) | (k & 1);
            unsigned ldsa = (unsigned)(uintptr_t)(lds_f + s);
            const float* ga = Wm + i;
            asm volatile("global_load_async_to_lds_b32 %0, %1, off"
                         :: "v"(ldsa), "v"(ga) : "memory");
        }
        asm volatile("s_wait_asynccnt 0x0" ::: "memory");
    }
    __syncthreads();

    const int wave = tid >> 5;
    const int lane = tid & 31;
    const int tm = blockIdx.x * (TPB / 32) + wave;     // 16-row tile index
    const int numTm = (N + 15) >> 4;
    if (tm >= numTm) return;                           // wave-uniform: EXEC stays full

    const int g = lane >> 4;                           // K-half select
    const int c = lane & 15;                           // row-in-tile (A) / col (B,C)

    int arow = tm * 16 + c;
    if (arow >= N) arow = N - 1;                       // clamp reads; stores guarded
    const float* ap = A + (size_t)arow * K;

    v8f acc0 = {}, acc1 = {}, acc2 = {}, acc3 = {};

    for (int k = 0; k < K; k += 4) {
        const int kb = k + 2 * g;
        v2f a;
        if (RELU) {
            a.x = fmaxf(ap[kb], 0.0f);
            a.y = fmaxf(ap[kb + 1], 0.0f);
        } else {
            a = *(const v2f*)(ap + kb);                // contiguous, 8B aligned
        }
        const v2f* wp = lwp + ((k >> 1) + g) * F;      // pair row for kb = k + 2g
        v2f fb0 = wp[c];
        v2f fb1 = wp[c + 16];
        v2f fb2 = wp[c + 32];
        v2f fb3 = wp[c + 48];
        acc0 = __builtin_amdgcn_wmma_f32_16x16x4_f32(false, a, false, fb0, (short)0, acc0, false, false);
        acc1 = __builtin_amdgcn_wmma_f32_16x16x4_f32(false, a, false, fb1, (short)0, acc1, false, false);
        acc2 = __builtin_amdgcn_wmma_f32_16x16x4_f32(false, a, false, fb2, (short)0, acc2, false, false);
        acc3 = __builtin_amdgcn_wmma_f32_16x16x4_f32(false, a, false, fb3, (short)0, acc3, false, false);
    }

    const float bs0 = bias[c];
    const float bs1 = bias[c + 16];
    const float bs2 = bias[c + 32];
    const float bs3 = bias[c + 48];

    float* hp = H   + (size_t)tm * 16 * F;
    float* gp = AGG + (size_t)tm * 16 * F;
#pragma unroll
    for (int v = 0; v < 8; ++v) {
        const int r = v + 8 * g;
        const int row = tm * 16 + r;
        if (row < N) {
            const float di = dinv[row];
            const float d2 = di * di;
            float* rp = hp + (size_t)r * F;
            float* ap2 = gp + (size_t)r * F;
            rp[c]       = acc0[v];  ap2[c]      = acc0[v] * d2 + bs0;
            rp[c + 16]  = acc1[v];  ap2[c + 16] = acc1[v] * d2 + bs1;
            rp[c + 32]  = acc2[v];  ap2[c + 32] = acc2[v] * d2 + bs2;
            rp[c + 48]  = acc3[v];  ap2[c + 48] = acc3[v] * d2 + bs3;
        }
    }
}

// One wave (32 lanes) per edge; each lane handles 2 of the 64 features.
// h/agg are L2-resident (25.6MB each vs 192MB L2), atomics hit L2.
__global__ void k_scatter(const float* __restrict__ h, const int* __restrict__ src,
                          const int* __restrict__ dst, const float* __restrict__ dinv,
                          float* __restrict__ agg, int E)
{
    long long gid = (long long)blockIdx.x * blockDim.x + threadIdx.x;
    int e = (int)(gid >> 5);
    if (e >= E) return;
    int l = (int)(gid & 31);
    int s = src[e], d = dst[e];
    float norm = dinv[s] * dinv[d];
    const float* hs = h + (size_t)s * 64;
    float* ad = agg + (size_t)d * 64;
    __hip_atomic_fetch_add(ad + l,      hs[l]      * norm,
                           __ATOMIC_RELAXED, __HIP_MEMORY_SCOPE_AGENT);
    __hip_atomic_fetch_add(ad + l + 32, hs[l + 32] * norm,
                           __ATOMIC_RELAXED, __HIP_MEMORY_SCOPE_AGENT);
}

extern "C" void kernel_launch(void* const* d_in, const int* in_sizes, int n_in,
                              void* d_out, int out_size, void* d_ws, size_t ws_size,
                              hipStream_t stream)
{
    const float* x  = (const float*)d_in[0];   // [N,128]
    const int*   ei = (const int*)d_in[1];     // [2,E]
    const float* W1 = (const float*)d_in[2];   // [128,64]
    const float* b1 = (const float*)d_in[3];   // [64]
    const float* W2 = (const float*)d_in[4];   // [64,64]
    const float* b2 = (const float*)d_in[5];   // [64]
    float* out = (float*)d_out;                // [N,64]

    const int N = in_sizes[0] / 128;
    const int E = in_sizes[1] / 2;
    const int* src = ei;
    const int* dst = ei + E;

    // Workspace carve-out (256B aligned): dinv[N], h1[N*64], agg1[N*64], h2[N*64]
    char* ws = (char*)d_ws;
    size_t off = 0;
    auto carve = [&](size_t bytes) {
        void* p = (void*)(ws + off);
        off = (off + bytes + 255) & ~(size_t)255;
        return p;
    };
    float* dinv = (float*)carve((size_t)N * 4);
    float* h1   = (float*)carve((size_t)N * 64 * 4);
    float* agg1 = (float*)carve((size_t)N * 64 * 4);
    float* h2   = (float*)carve((size_t)N * 64 * 4);

    dim3 blk(TPB);
    dim3 gN((N + TPB - 1) / TPB);
    dim3 gE((E + TPB - 1) / TPB);
    dim3 gEdgeWave((unsigned)(((long long)E * 32 + TPB - 1) / TPB));
    const int numTm = (N + 15) / 16;
    dim3 gGemm((numTm + (TPB / 32) - 1) / (TPB / 32));

    // Degrees (with self loop) -> dinv = rsqrt(deg)
    k_fill<<<gN, blk, 0, stream>>>(dinv, 1.0f, N);
    k_deg_count<<<gE, blk, 0, stream>>>(dst, dinv, E);
    k_rsqrt<<<gN, blk, 0, stream>>>(dinv, N);

    // Layer 1: h1 = x @ W1 ; agg1 = self + b1 (fused) ; then edge scatter
    k_gemm_wmma<false, 128><<<gGemm, blk, 0, stream>>>(x, W1, h1, agg1, dinv, b1, N);
    k_scatter<<<gEdgeWave, blk, 0, stream>>>(h1, src, dst, dinv, agg1, E);

    // Layer 2: h2 = relu(agg1) @ W2 ; out = self + b2 (fused) ; then edge scatter
    k_gemm_wmma<true, 64><<<gGemm, blk, 0, stream>>>(agg1, W2, h2, out, dinv, b2, N);
    k_scatter<<<gEdgeWave, blk, 0, stream>>>(h2, src, dst, dinv, out, E);
}